// YOLOLoss_84963043049786
// MI455X (gfx1250) — compile-verified
//
#include <hip/hip_runtime.h>

// ---------------- configuration ----------------
#define THREADS     256
#define CPB         256                    // cells per block
#define CELLF       30                     // floats per cell
#define TILE_FLOATS (CPB * CELLF)          // 7680 floats per tensor tile
#define TILE_BYTES  (TILE_FLOATS * 4)      // 30720 B per tensor tile
#define STAGE_ITERS ((TILE_FLOATS / 4 + THREADS - 1) / THREADS)  // 8 (1920 chunks)

#if defined(__HIP_DEVICE_COMPILE__) && __has_builtin(__builtin_amdgcn_global_load_async_to_lds_b128)
#define HAVE_ASYNC_LDS 1
#else
#define HAVE_ASYNC_LDS 0
#endif

// GCC-style int4 vector to match the builtin's parameter type exactly
typedef int v4i_t __attribute__((vector_size(16)));
typedef __attribute__((address_space(1))) v4i_t* glob_v4i_ptr;
typedef __attribute__((address_space(3))) v4i_t* lds_v4i_ptr;

// ---------------- staging helpers ----------------
__device__ __forceinline__ void stage16(const float* __restrict__ g, float* l) {
#if HAVE_ASYNC_LDS
  // async DMA: global -> LDS, 16 bytes per lane, tracked by ASYNCcnt
  __builtin_amdgcn_global_load_async_to_lds_b128(
      (glob_v4i_ptr)(v4i_t*)g,
      (lds_v4i_ptr)(v4i_t*)l,
      /*imm offset*/ 0, /*cpol*/ 0);
#else
  *(float4*)l = *(const float4*)g;
#endif
}

__device__ __forceinline__ void wait_async_done() {
#if HAVE_ASYNC_LDS
#if __has_builtin(__builtin_amdgcn_s_wait_asynccnt)
  __builtin_amdgcn_s_wait_asynccnt(0);
#else
  asm volatile("s_wait_asynccnt 0" ::: "memory");
#endif
#endif
}

// ---------------- per-cell loss (matches reference exactly) ----------------
__device__ __forceinline__ float cell_loss(const float* __restrict__ p,
                                           const float* __restrict__ l) {
  const float S_INV = 1.0f / 7.0f;

  if (!(l[4] > 0.0f)) {
    // no-object cell: NOOBJ * sum over both boxes of (p_conf - l_conf)^2
    float d0 = p[4] - l[4];
    float d1 = p[9] - l[9];
    return 0.5f * (d0 * d0 + d1 * d1);
  }

  // ground-truth corners from label box 0 (computed corner-style like reference)
  float gx = l[0] * S_INV, gy = l[1] * S_INV;
  float gx1 = gx - 0.5f * l[2], gy1 = gy - 0.5f * l[3];
  float gx2 = gx + 0.5f * l[2], gy2 = gy + 0.5f * l[3];
  float ga  = (gx2 - gx1) * (gy2 - gy1);

  float iou0 = 0.0f, iou1 = 0.0f;
#pragma unroll
  for (int b = 0; b < 2; ++b) {
    const float* bp = p + 5 * b;
    float px = bp[0] * S_INV, py = bp[1] * S_INV;
    float x1 = px - 0.5f * bp[2], y1 = py - 0.5f * bp[3];
    float x2 = px + 0.5f * bp[2], y2 = py + 0.5f * bp[3];
    float ix1 = fmaxf(x1, gx1), iy1 = fmaxf(y1, gy1);
    float ix2 = fminf(x2, gx2), iy2 = fminf(y2, gy2);
    float inter = fmaxf(ix2 - ix1, 0.0f) * fmaxf(iy2 - iy1, 0.0f);
    float a1 = (x2 - x1) * (y2 - y1);
    float iou = inter / (a1 + ga - inter);
    if (b == 0) iou0 = iou; else iou1 = iou;
  }

  // argmax picks first occurrence of max -> box 0 on ties
  int   best    = (iou1 > iou0) ? 1 : 0;
  float max_iou = fmaxf(iou0, iou1);

  const float* rp  = p + 5 * best;     // responsible pred box
  const float* rl  = l + 5 * best;     // matching label box
  const float* np_ = p + 5 * (1 - best);

  float dx = rp[0] - rl[0], dy = rp[1] - rl[1];
  float xy = dx * dx + dy * dy;

  float dw = sqrtf(rp[2]) - sqrtf(rl[2]);
  float dh = sqrtf(rp[3]) - sqrtf(rl[3]);
  float wh = dw * dw + dh * dh;

  float dr   = rp[4] - max_iou;
  float resp = dr * dr;
  float nr   = np_[4] * np_[4];

  float cls = 0.0f;
#pragma unroll
  for (int k = 10; k < 30; ++k) {
    float d = p[k] - l[k];
    cls += d * d;
  }

  return 5.0f * (xy + wh) + 2.0f * resp + nr + cls;
}

// ---------------- main kernel: stage tile via async DMA, compute, block-reduce ----------------
__global__ void yolo_loss_main(const float* __restrict__ preds,
                               const float* __restrict__ labels,
                               float* __restrict__ partials,
                               int totalCells) {
  extern __shared__ float smem[];
  float* sp = smem;               // staged preds tile  (TILE_FLOATS)
  float* sl = smem + TILE_FLOATS; // staged labels tile (TILE_FLOATS)

  const int t = threadIdx.x;
  const int blockBase = blockIdx.x * CPB;
  int cellsHere = totalCells - blockBase;
  if (cellsHere > CPB) cellsHere = CPB;
  if (cellsHere < 0)   cellsHere = 0;

  const size_t fbase = (size_t)blockBase * CELLF;
  const float* gp = preds  + fbase;
  const float* gl = labels + fbase;

  const int floatsTile = cellsHere * CELLF;
  const int fullChunks = floatsTile >> 2;     // # of complete 16B chunks

#pragma unroll
  for (int i = 0; i < STAGE_ITERS; ++i) {
    int c = i * THREADS + t;
    if (c < fullChunks) {
      stage16(gp + 4 * c, sp + 4 * c);
      stage16(gl + 4 * c, sl + 4 * c);
    }
  }
  // tail floats (only when cellsHere*30 % 4 != 0, i.e. partial last tile)
  if (t < (floatsTile & 3)) {
    int f = (fullChunks << 2) + t;
    sp[f] = gp[f];
    sl[f] = gl[f];
  }

  wait_async_done();
  __syncthreads();

  float acc = 0.0f;
  if (t < cellsHere)
    acc = cell_loss(sp + CELLF * t, sl + CELLF * t);

  // wave32 reduction
#pragma unroll
  for (int off = 16; off > 0; off >>= 1)
    acc += __shfl_xor(acc, off, 32);

  __shared__ float red[THREADS / 32];
  const int lane = t & 31;
  const int w    = t >> 5;
  if (lane == 0) red[w] = acc;
  __syncthreads();

  if (t == 0) {
    float s = 0.0f;
#pragma unroll
    for (int i = 0; i < THREADS / 32; ++i) s += red[i];
    partials[blockIdx.x] = s;
  }
}

// ---------------- finalize: deterministic sum of block partials ----------------
__global__ void yolo_loss_finalize(const float* __restrict__ partials,
                                   int nParts, float* __restrict__ out,
                                   float invN) {
  const int t = threadIdx.x;
  float s = 0.0f;
  for (int i = t; i < nParts; i += THREADS) s += partials[i];

#pragma unroll
  for (int off = 16; off > 0; off >>= 1)
    s += __shfl_xor(s, off, 32);

  __shared__ float red[THREADS / 32];
  const int lane = t & 31;
  const int w    = t >> 5;
  if (lane == 0) red[w] = s;
  __syncthreads();

  if (t == 0) {
    float tot = 0.0f;
#pragma unroll
    for (int i = 0; i < THREADS / 32; ++i) tot += red[i];
    out[0] = tot * invN;
  }
}

// ---------------- host entry ----------------
extern "C" void kernel_launch(void* const* d_in, const int* in_sizes, int n_in,
                              void* d_out, int out_size, void* d_ws, size_t ws_size,
                              hipStream_t stream) {
  const float* preds  = (const float*)d_in[0];
  const float* labels = (const float*)d_in[1];

  const int totalFloats = in_sizes[0];                 // 16384*7*7*30
  const int totalCells  = totalFloats / CELLF;         // 802816
  const int batchN      = totalFloats / (49 * CELLF);  // 16384
  const int nBlocks     = (totalCells + CPB - 1) / CPB; // 3136

  float* partials = (float*)d_ws;
  const size_t shmem = (size_t)2 * TILE_BYTES;         // 61440 B

  yolo_loss_main<<<nBlocks, THREADS, shmem, stream>>>(preds, labels, partials, totalCells);
  yolo_loss_finalize<<<1, THREADS, 0, stream>>>(partials, nBlocks, (float*)d_out,
                                                1.0f / (float)batchN);
}